// MultiScaleWaveletNet_36747740185144
// MI455X (gfx1250) — compile-verified
//
#include <hip/hip_runtime.h>
#include <hip/hip_bf16.h>

// MI455X / gfx1250 MultiScaleWaveletNet.
// - fp32 WMMA (V_WMMA_F32_16X16X4_F32) for the implicit-GEMM 3x3 convs and the
//   final 1x1 conv GEMM (memory-bound at 23.3 TB/s, so fp32 matrix rate is free).
// - Weights staged to LDS via async global->LDS copies (ASYNCcnt) when the
//   toolchain exposes the gfx1250 builtins.
// - 3x3 conv K-decode is fully constant-folded (template HL, literal tap
//   offsets selected by one cndmask on the lane half), and boundary handling is
//   a wave-uniform branch so EXEC is all-ones at every WMMA.
// - Biases skipped: instance norm subtracts the per-plane mean, so per-channel
//   bias cancels exactly.

typedef float v2f __attribute__((ext_vector_type(2)));
typedef float v8f __attribute__((ext_vector_type(8)));

#define EPSV 1e-5f

#if defined(__has_builtin)
#if __has_builtin(__builtin_amdgcn_global_load_async_to_lds_b32) && \
    __has_builtin(__builtin_amdgcn_s_wait_asynccnt)
#define USE_ASYNC_LDS 1
#endif
#endif

#ifdef USE_ASYNC_LDS
// HIP prints addrspace(1) as "__device__" and addrspace(3) as "__shared__";
// the builtin wants (addrspace(1) int*, addrspace(3) int*, imm, imm).
typedef __attribute__((address_space(1))) int gbl_i;
typedef __attribute__((address_space(3))) int lds_i;
#endif

// Cooperative block-wide copy of n floats global->LDS.
__device__ __forceinline__ void stage_to_lds(const float* __restrict__ g,
                                             float* l, int n) {
#ifdef USE_ASYNC_LDS
  for (int i = threadIdx.x; i < n; i += blockDim.x) {
    __builtin_amdgcn_global_load_async_to_lds_b32(
        (gbl_i*)(g + i), (lds_i*)(l + i), 0, 0);
  }
  __builtin_amdgcn_s_wait_asynccnt(0);
#else
  for (int i = threadIdx.x; i < n; i += blockDim.x) l[i] = g[i];
#endif
  __syncthreads();
}

__device__ __forceinline__ v8f wmma_k4(v2f a, v2f b, v8f c) {
  // D = A(16x4 f32) * B(4x16 f32) + C(16x16 f32)
  return __builtin_amdgcn_wmma_f32_16x16x4_f32(
      /*neg_a=*/false, a, /*neg_b=*/false, b,
      /*c_mod=*/(short)0, c, /*reuse_a=*/false, /*reuse_b=*/false);
}

// ---------------------------------------------------------------- avg pool 2x2
__global__ void avgpool2_kernel(const float* __restrict__ in, float* __restrict__ out,
                                int BC, int Hin, int Win) {
  int Ho = Hin >> 1, Wo = Win >> 1;
  long idx = (long)blockIdx.x * blockDim.x + threadIdx.x;
  int x = (int)(idx % Wo); long t = idx / Wo;
  int y = (int)(t % Ho);   int bc = (int)(t / Ho);
  const float* p = in + ((long)bc * Hin + 2 * y) * Win + 2 * x;
  out[idx] = 0.25f * (p[0] + p[1] + p[Win] + p[Win + 1]);
}

// ------------------------------------------- Haar bands + bilinear x2 upsample
// cur: (B, C, Hl, Wl)  ->  out: (B, 4*C, Hl, Wl); band grid is (Hl/2, Wl/2).
__global__ void haar_up_kernel(const float* __restrict__ cur, float* __restrict__ out,
                               int C, int Hl, int Wl) {
  long idx = (long)blockIdx.x * blockDim.x + threadIdx.x;
  int x = (int)(idx % Wl); long t = idx / Wl;
  int y = (int)(t % Hl);   t /= Hl;
  int c4 = (int)(t % (4 * C));
  int b  = (int)(t / (4 * C));
  int c = c4 >> 2, band = c4 & 3;
  int Hh = Hl >> 1, Wh = Wl >> 1;
  float sF = (band & 2) ? -1.0f : 1.0f;  // sign of odd-col sample
  float sG = (band & 1) ? -1.0f : 1.0f;  // sign of odd-row sample
  float sH = sF * sG;
  // half-pixel bilinear: src = (i+0.5)/2 - 0.5
  float sy = 0.5f * (float)y - 0.25f;
  float sx = 0.5f * (float)x - 0.25f;
  int y0 = (int)floorf(sy); float wy = sy - (float)y0;
  int x0 = (int)floorf(sx); float wx = sx - (float)x0;
  int iy0 = y0 < 0 ? 0 : y0;
  int iy1 = (y0 + 1 > Hh - 1) ? Hh - 1 : y0 + 1;
  int ix0 = x0 < 0 ? 0 : x0;
  int ix1 = (x0 + 1 > Wh - 1) ? Wh - 1 : x0 + 1;
  const float* p = cur + ((long)b * C + c) * Hl * Wl;
  auto bandval = [&](int i, int j) -> float {
    const float* q = p + (long)(2 * i) * Wl + 2 * j;
    float e = q[0], f = q[1], g = q[Wl], h = q[Wl + 1];
    return 0.5f * (e + sF * f + sG * g + sH * h);
  };
  float v = (1.0f - wy) * ((1.0f - wx) * bandval(iy0, ix0) + wx * bandval(iy0, ix1)) +
            wy          * ((1.0f - wx) * bandval(iy1, ix0) + wx * bandval(iy1, ix1));
  out[idx] = v;
}

// ------------------------------------- 3x3 conv (SAME) as implicit-GEMM WMMA
// bands: (B, 12, HL, HL), w: (32, 12, 3, 3) flat [co*108 + c*9 + ky*3 + kx]
// One wave: 32(co) x 16(pixel) tile, K = 108 in steps of 4.
// A = weights (M=co, from LDS), B = im2col pixels (N=pixel).
// Tap offsets are compile-time literals; lane half (hi) picks K or K+2 via one
// cndmask. Interior/edge split is wave-uniform -> EXEC all-ones at every WMMA.
template <int HL, bool EDGE>
__device__ __forceinline__ void conv_k_loop(const float* __restrict__ pb,
                                            const float* __restrict__ sw,
                                            int nsub, int hi, int y, int xg,
                                            v8f& acc0, v8f& acc1) {
  bool vy[3] = { y > 0, true, y < HL - 1 };
  bool vx[3] = { xg > 0, true, xg < HL - 1 };
#pragma unroll
  for (int kb = 0; kb < 108; kb += 4) {
    v2f bf;
#pragma unroll
    for (int j = 0; j < 2; ++j) {
      const int kA = kb + j;        // lanes 0-15  (hi == 0)
      const int kB = kb + 2 + j;    // lanes 16-31 (hi == 1)
      const int cA = kA / 9, rA = kA % 9, kyA = rA / 3, kxA = rA % 3;
      const int cB = kB / 9, rB = kB % 9, kyB = rB / 3, kxB = rB % 3;
      const int offA = (cA * HL + (kyA - 1)) * HL + (kxA - 1);
      const int offB = (cB * HL + (kyB - 1)) * HL + (kxB - 1);
      int off = hi ? offB : offA;   // one v_cndmask of two inline constants
      float v;
      if (EDGE) {
        bool ok = hi ? (vy[kyB] && vx[kxB]) : (vy[kyA] && vx[kxA]);
        v = pb[ok ? off : 0] * (ok ? 1.0f : 0.0f);  // branchless, in-bounds
      } else {
        v = pb[off];
      }
      if (j == 0) bf.x = v; else bf.y = v;
    }
    int k0 = kb + (hi << 1);
    v2f a0 = *(const v2f*)(sw + nsub * 108 + k0);          // ds_load_b64
    v2f a1 = *(const v2f*)(sw + (16 + nsub) * 108 + k0);   // ds_load_b64
    acc0 = wmma_k4(a0, bf, acc0);
    acc1 = wmma_k4(a1, bf, acc1);
  }
}

template <int HL>
__global__ __launch_bounds__(256)
void conv3x3_wmma_kernel(const float* __restrict__ bands,
                         const float* __restrict__ w,
                         float* __restrict__ outb,
                         long out_bstride, long out_cstride) {
  __shared__ float sw[32 * 108];
  stage_to_lds(w, sw, 32 * 108);
  const int lane = threadIdx.x & 31;
  const int nsub = lane & 15;
  const int hi   = lane >> 4;
  int wid = blockIdx.x * (blockDim.x >> 5) + (threadIdx.x >> 5);
  constexpr int tilesX = HL >> 4;
  int tx = wid % tilesX; int rest = wid / tilesX;
  int y = rest % HL;     int b = rest / HL;
  int xg = (tx << 4) + nsub;          // this lane's pixel column (B/C N index)
  const float* pb = bands + (long)b * 12 * HL * HL + y * HL + xg;
  v8f acc0 = {}; v8f acc1 = {};
  bool edge = (y == 0) || (y == HL - 1) || (tx == 0) || (tx == tilesX - 1);
  if (edge) {                          // wave-uniform branch
    conv_k_loop<HL, true >(pb, sw, nsub, hi, y, xg, acc0, acc1);
  } else {
    conv_k_loop<HL, false>(pb, sw, nsub, hi, y, xg, acc0, acc1);
  }
  long obase = (long)b * out_bstride + (long)y * HL + xg;
#pragma unroll
  for (int r = 0; r < 8; ++r) {
    int m = r + (hi << 3);             // C/D row = co
    outb[obase + (long)m * out_cstride]        = acc0[r];
    outb[obase + (long)(16 + m) * out_cstride] = acc1[r];
  }
}

// --------------------------------------------------- instance-norm statistics
__global__ void instnorm_stats_kernel(const float* __restrict__ base,
                                      long bstride, long cstride,
                                      int C, int npix, float* __restrict__ stats) {
  __shared__ float ssum[256];
  __shared__ float ssq[256];
  int bc = blockIdx.x;
  int b = bc / C, c = bc % C;
  const float* p = base + (long)b * bstride + (long)c * cstride;
  float s = 0.0f, q = 0.0f;
  for (int i = threadIdx.x; i < npix; i += blockDim.x) {
    float v = p[i]; s += v; q += v * v;
  }
  ssum[threadIdx.x] = s; ssq[threadIdx.x] = q;
  __syncthreads();
  for (int off = 128; off > 0; off >>= 1) {
    if ((int)threadIdx.x < off) {
      ssum[threadIdx.x] += ssum[threadIdx.x + off];
      ssq[threadIdx.x]  += ssq[threadIdx.x + off];
    }
    __syncthreads();
  }
  if (threadIdx.x == 0) {
    float mean = ssum[0] / (float)npix;
    float var = ssq[0] / (float)npix - mean * mean;
    var = var < 0.0f ? 0.0f : var;
    stats[2 * bc]     = mean;
    stats[2 * bc + 1] = rsqrtf(var + EPSV);
  }
}

// --------------------- normalize + PReLU, then bilinear upsample into concat
// raw: 32 channels at Hin x Hin; writes cat[b, ch_off+c, 256, 256].
__global__ void norm_prelu_up_kernel(const float* __restrict__ raw,
                                     long r_bstride, long r_cstride, int Hin,
                                     const float* __restrict__ stats,
                                     const float* __restrict__ aptr,
                                     float* __restrict__ cat, int ch_off) {
  unsigned idx = blockIdx.x * blockDim.x + threadIdx.x;   // < 2^25
  int x = idx & 255, y = (idx >> 8) & 255, c = (idx >> 16) & 31, b = idx >> 21;
  float alpha = aptr[0];
  float mean = stats[2 * (b * 32 + c)];
  float rstd = stats[2 * (b * 32 + c) + 1];
  const float* p = raw + (long)b * r_bstride + (long)c * r_cstride;
  float v;
  if (Hin == 256) {                      // level 0: no resize, in-place safe
    float tv = (p[(y << 8) + x] - mean) * rstd;
    v = tv < 0.0f ? alpha * tv : tv;
  } else {
    float sc = (float)Hin * (1.0f / 256.0f);
    float sy = ((float)y + 0.5f) * sc - 0.5f;
    float sx = ((float)x + 0.5f) * sc - 0.5f;
    int y0 = (int)floorf(sy); float wy = sy - (float)y0;
    int x0 = (int)floorf(sx); float wx = sx - (float)x0;
    int iy0 = y0 < 0 ? 0 : y0;
    int iy1 = (y0 + 1 > Hin - 1) ? Hin - 1 : y0 + 1;
    int ix0 = x0 < 0 ? 0 : x0;
    int ix1 = (x0 + 1 > Hin - 1) ? Hin - 1 : x0 + 1;
    auto np = [&](int yy, int xx) -> float {   // PReLU BEFORE resize (ref order)
      float tv = (p[yy * Hin + xx] - mean) * rstd;
      return tv < 0.0f ? alpha * tv : tv;
    };
    v = (1.0f - wy) * ((1.0f - wx) * np(iy0, ix0) + wx * np(iy0, ix1)) +
        wy          * ((1.0f - wx) * np(iy1, ix0) + wx * np(iy1, ix1));
  }
  cat[(((long)b * 96 + ch_off + c) << 16) + (y << 8) + x] = v;
}

// ------------------------------------------ final 1x1 conv as WMMA GEMM
// out_raw[b, co, p] = sum_k wf[co,k] * cat[b,k,p]; M=co(96, 6 tiles), N=16
// pixels, K=96 (24 steps of 4). Weights in LDS; prefetch next cat rows.
__global__ __launch_bounds__(256)
void final_conv_wmma_kernel(const float* __restrict__ cat,
                            const float* __restrict__ wf,
                            float* __restrict__ out) {
  __shared__ float sw[96 * 96];
  stage_to_lds(wf, sw, 96 * 96);
  const int lane = threadIdx.x & 31;
  const int nsub = lane & 15;
  const int hi   = lane >> 4;
  int wid = blockIdx.x * (blockDim.x >> 5) + (threadIdx.x >> 5);
  int b  = wid >> 12;                 // 4096 pixel-tiles of 16 per batch image
  int pt = (wid & 4095) << 4;
  const float* cb = cat + (long)b * 96 * 65536 + pt + nsub;
  v8f acc[6] = {{}, {}, {}, {}, {}, {}};
#pragma unroll
  for (int kb = 0; kb < 96; kb += 4) {
    int k0 = kb + (hi << 1);
    v2f bf;
    bf.x = cb[(long)k0 * 65536];
    bf.y = cb[(long)(k0 + 1) * 65536];
    if (kb + 4 < 96)                  // uniform branch; speculative L2 prefetch
      __builtin_prefetch(cb + (long)(kb + 4 + (hi << 1)) * 65536, 0, 0);
#pragma unroll
    for (int m = 0; m < 6; ++m) {
      v2f a = *(const v2f*)(sw + (m * 16 + nsub) * 96 + k0);  // ds_load_b64
      acc[m] = wmma_k4(a, bf, acc[m]);
    }
  }
  long obase = (long)b * 96 * 65536 + pt + nsub;
#pragma unroll
  for (int m = 0; m < 6; ++m) {
#pragma unroll
    for (int r = 0; r < 8; ++r) {
      out[obase + (long)(m * 16 + r + (hi << 3)) * 65536] = acc[m][r];
    }
  }
}

// ------------------------------------------ final IN + PReLU, in place on out
__global__ void final_norm_prelu_kernel(float* __restrict__ out,
                                        const float* __restrict__ stats,
                                        const float* __restrict__ aptr) {
  long idx = (long)blockIdx.x * blockDim.x + threadIdx.x;  // < 2^27
  int bc = (int)(idx >> 16);
  float mean = stats[2 * bc], rstd = stats[2 * bc + 1];
  float alpha = aptr[0];
  float tv = (out[idx] - mean) * rstd;
  out[idx] = tv < 0.0f ? alpha * tv : tv;
}

// ============================================================================
extern "C" void kernel_launch(void* const* d_in, const int* in_sizes, int n_in,
                              void* d_out, int out_size, void* d_ws, size_t ws_size,
                              hipStream_t stream) {
  const float* x  = (const float*)d_in[0];
  const float* w0 = (const float*)d_in[1];
  const float* a0 = (const float*)d_in[3];
  const float* w1 = (const float*)d_in[4];
  const float* a1 = (const float*)d_in[6];
  const float* w2 = (const float*)d_in[7];
  const float* a2 = (const float*)d_in[9];
  const float* wf = (const float*)d_in[10];
  const float* af = (const float*)d_in[12];
  float* out = (float*)d_out;

  // workspace layout (floats); raw1/raw2 alias dead band buffers
  const long N_X1 = 16L * 3 * 128 * 128;     //   786432
  const long N_X2 = 16L * 3 * 64 * 64;       //   196608
  const long N_B0 = 16L * 12 * 256 * 256;    // 12582912 (>= raw1: 8388608)
  const long N_B1 = 16L * 12 * 128 * 128;    //  3145728 (>= raw2: 2097152)
  const long N_B2 = 16L * 12 * 64 * 64;      //   786432
  const long N_CAT = 16L * 96 * 256 * 256;   // 100663296
  float* ws    = (float*)d_ws;
  float* x1    = ws;
  float* x2    = x1 + N_X1;
  float* b0buf = x2 + N_X2;
  float* b1buf = b0buf + N_B0;
  float* b2buf = b1buf + N_B1;
  float* cat   = b2buf + N_B2;
  float* stL   = cat + N_CAT;     // 1024 floats, reused per level
  float* stF   = stL + 1024;      // 3072 floats
  float* raw1  = b0buf;
  float* raw2  = b1buf;

  const dim3 T(256);
  const long CAT_B = 96L * 65536;   // cat/out batch stride
  const long CAT_C = 65536;

  // pyramid
  avgpool2_kernel<<<dim3(3072), T, 0, stream>>>(x, x1, 48, 256, 256);
  avgpool2_kernel<<<dim3(768),  T, 0, stream>>>(x1, x2, 48, 128, 128);

  // ---- level 0 (raw conv output lives directly in cat channels 0..31)
  haar_up_kernel<<<dim3(49152), T, 0, stream>>>(x, b0buf, 3, 256, 256);
  conv3x3_wmma_kernel<256><<<dim3(8192), T, 0, stream>>>(b0buf, w0, cat, CAT_B, CAT_C);
  instnorm_stats_kernel<<<dim3(512), T, 0, stream>>>(cat, CAT_B, CAT_C, 32, 65536, stL);
  norm_prelu_up_kernel<<<dim3(131072), T, 0, stream>>>(cat, CAT_B, CAT_C, 256, stL, a0, cat, 0);

  // ---- level 1
  haar_up_kernel<<<dim3(12288), T, 0, stream>>>(x1, b1buf, 3, 128, 128);
  conv3x3_wmma_kernel<128><<<dim3(2048), T, 0, stream>>>(b1buf, w1, raw1,
                                                         32L * 16384, 16384L);
  instnorm_stats_kernel<<<dim3(512), T, 0, stream>>>(raw1, 32L * 16384, 16384L, 32, 16384, stL);
  norm_prelu_up_kernel<<<dim3(131072), T, 0, stream>>>(raw1, 32L * 16384, 16384L, 128, stL, a1, cat, 32);

  // ---- level 2
  haar_up_kernel<<<dim3(3072), T, 0, stream>>>(x2, b2buf, 3, 64, 64);
  conv3x3_wmma_kernel<64><<<dim3(512), T, 0, stream>>>(b2buf, w2, raw2,
                                                       32L * 4096, 4096L);
  instnorm_stats_kernel<<<dim3(512), T, 0, stream>>>(raw2, 32L * 4096, 4096L, 32, 4096, stL);
  norm_prelu_up_kernel<<<dim3(131072), T, 0, stream>>>(raw2, 32L * 4096, 4096L, 64, stL, a2, cat, 64);

  // ---- final 1x1 conv GEMM + IN + PReLU
  final_conv_wmma_kernel<<<dim3(8192), T, 0, stream>>>(cat, wf, out);
  instnorm_stats_kernel<<<dim3(1536), T, 0, stream>>>(out, CAT_B, CAT_C, 96, 65536, stF);
  final_norm_prelu_kernel<<<dim3(393216), T, 0, stream>>>(out, stF, af);
}